// GIN_60155311948561
// MI455X (gfx1250) — compile-verified
//
#include <hip/hip_runtime.h>

#define N_NODES   100000
#define N_EDGES   1600000
#define DIM       128
#define N_LAYERS  3
#define N_GRAPHS  128
#define N_CLASSES 10
#define BN_EPS    1e-5f

static const size_t ND = (size_t)N_NODES * DIM;

typedef __attribute__((ext_vector_type(2))) float v2f;
typedef __attribute__((ext_vector_type(8))) float v8f;

// ---------------------------------------------------------------- utilities
__global__ void zero_f32(float* __restrict__ p, int n) {
  int i = blockIdx.x * blockDim.x + threadIdx.x;
  if (i < n) p[i] = 0.0f;
}

// Wt[n][k] = W[k][n]  for a 128x128 weight (so B fragments load as one b64)
__global__ void transpose128(const float* __restrict__ W, float* __restrict__ Wt) {
  int i = blockIdx.x * blockDim.x + threadIdx.x;   // 16384
  int n = i >> 7;
  int k = i & (DIM - 1);
  Wt[i] = W[(size_t)k * DIM + n];
}

// ---------------------------------------------------------------- GIN aggregation
// agg[i] = (1 + eps[layer]) * x[i]   (vectorized float4, exact grid)
__global__ void gin_init(const float* __restrict__ x, float* __restrict__ agg,
                         const float* __restrict__ eps, int layer) {
  size_t i = (size_t)blockIdx.x * blockDim.x + threadIdx.x;   // float4 index
  float s = 1.0f + eps[layer];
  const float4* __restrict__ x4 = (const float4*)x;
  float4* __restrict__ a4 = (float4*)agg;
  float4 v = x4[i];
  v.x *= s; v.y *= s; v.z *= s; v.w *= s;
  a4[i] = v;
}

// agg[dst[e]] += x[src[e]]   one wave per edge, lane = one float4 (16 B)
__global__ void gin_scatter(const float* __restrict__ x,
                            const int* __restrict__ src,
                            const int* __restrict__ dst,
                            float* __restrict__ agg, int nEdges) {
  int wavesPerBlock = blockDim.x >> 5;
  int e    = blockIdx.x * wavesPerBlock + (threadIdx.x >> 5);
  int lane = threadIdx.x & 31;
  if (e >= nEdges) return;
  int s = src[e];
  int d = dst[e];
  const float4* __restrict__ xs = (const float4*)(x + (size_t)s * DIM);
  float4 v = xs[lane];
  float* __restrict__ ad = agg + (size_t)d * DIM + lane * 4;
  atomicAdd(ad + 0, v.x);
  atomicAdd(ad + 1, v.y);
  atomicAdd(ad + 2, v.z);
  atomicAdd(ad + 3, v.w);
}

// ---------------------------------------------------------------- WMMA f32 GEMM
// O[M x 128] = relu(A[M x 128] @ W[128 x 128] + bias), W given TRANSPOSED (Wt[n][k]).
// grid.x = M/16 (row tiles), block = 256 threads = 8 waves, wave w -> col tile w.
// EXEC is all-ones everywhere (M is a multiple of 16), as WMMA requires.
// Inner loop: one b64 A load + one b64 B load + one v_wmma_f32_16x16x4_f32.
__global__ void gemm_bias_relu(const float* __restrict__ A,
                               const float* __restrict__ Wt,
                               const float* __restrict__ bias,
                               float* __restrict__ O) {
  int wave = threadIdx.x >> 5;       // 0..7  -> N tile
  int lane = threadIdx.x & 31;
  int m0 = blockIdx.x * 16;
  int n0 = wave * 16;
  int mrow = lane & 15;              // A-fragment row
  int hi   = lane >> 4;              // lane-half selects K offset (+2)
  int ncol = lane & 15;              // B/C/D column within tile

  v8f c = {};
  const v2f* __restrict__ A2 = (const v2f*)(A  + (size_t)(m0 + mrow) * DIM);
  const v2f* __restrict__ B2 = (const v2f*)(Wt + (size_t)(n0 + ncol) * DIM);
#pragma unroll
  for (int k0 = 0; k0 < DIM; k0 += 4) {
    // A 16x4 f32 layout: VGPR0 = K=k0+2*hi, VGPR1 = K=k0+2*hi+1 (ISA 7.12.2)
    v2f a = A2[(k0 >> 1) + hi];
    // B 4x16 f32 layout mirrors A; Wt makes (W[k][n], W[k+1][n]) contiguous
    v2f b = B2[(k0 >> 1) + hi];
    c = __builtin_amdgcn_wmma_f32_16x16x4_f32(
        /*neg_a=*/false, a, /*neg_b=*/false, b,
        /*c_mod=*/(short)0, c, /*reuse_a=*/false, /*reuse_b=*/false);
  }

  float bv = bias[n0 + ncol];
#pragma unroll
  for (int r = 0; r < 8; ++r) {      // C/D: VGPR r -> M = r + 8*hi, N = ncol
    float v = c[r] + bv;
    v = v > 0.0f ? v : 0.0f;
    O[(size_t)(m0 + r + 8 * hi) * DIM + n0 + ncol] = v;
  }
}

// ---------------------------------------------------------------- BatchNorm
#define BN_ROWS 500   // 100000 / 500 = 200 blocks
__global__ void bn_reduce(const float* __restrict__ h, float* __restrict__ stats, int n) {
  int d  = threadIdx.x;              // 128 threads: column d (coalesced)
  int r0 = blockIdx.x * BN_ROWS;
  float s = 0.0f, s2 = 0.0f;
  int rEnd = r0 + BN_ROWS; if (rEnd > n) rEnd = n;
  for (int r = r0; r < rEnd; ++r) {
    float v = h[(size_t)r * DIM + d];
    s += v; s2 += v * v;
  }
  atomicAdd(&stats[d], s);
  atomicAdd(&stats[DIM + d], s2);
}

__global__ void bn_norm(const float* __restrict__ h, const float* __restrict__ stats,
                        const float* __restrict__ gamma, const float* __restrict__ beta,
                        float* __restrict__ out, int n) {
  size_t i = (size_t)blockIdx.x * blockDim.x + threadIdx.x;
  int d = (int)(i & (DIM - 1));
  float inv_n = 1.0f / (float)n;
  float mu  = stats[d] * inv_n;
  float var = stats[DIM + d] * inv_n - mu * mu;
  float sc  = gamma[d] * rsqrtf(var + BN_EPS);
  out[i] = sc * (h[i] - mu) + beta[d];
}

// ---------------------------------------------------------------- global mean pool
__global__ void pool_scatter(const float* __restrict__ x, const int* __restrict__ batch,
                             float* __restrict__ sums, float* __restrict__ cnts, int n) {
  int wavesPerBlock = blockDim.x >> 5;
  int i    = blockIdx.x * wavesPerBlock + (threadIdx.x >> 5);
  int lane = threadIdx.x & 31;
  if (i >= n) return;
  int g = batch[i];
  const float4* __restrict__ xs = (const float4*)(x + (size_t)i * DIM);
  float4 v = xs[lane];
  float* __restrict__ sd = sums + (size_t)g * DIM + lane * 4;
  atomicAdd(sd + 0, v.x);
  atomicAdd(sd + 1, v.y);
  atomicAdd(sd + 2, v.z);
  atomicAdd(sd + 3, v.w);
  if (lane == 0) atomicAdd(&cnts[g], 1.0f);
}

__global__ void pool_div(const float* __restrict__ sums, const float* __restrict__ cnts,
                         float* __restrict__ pooled) {
  int i = blockIdx.x * blockDim.x + threadIdx.x;   // 128*128
  if (i >= N_GRAPHS * DIM) return;
  int g = i >> 7;
  float c = cnts[g];
  pooled[i] = sums[i] / (c > 1.0f ? c : 1.0f);
}

// ---------------------------------------------------------------- head: fc2 + log_softmax
__global__ void head_kernel(const float* __restrict__ ph, const float* __restrict__ W2,
                            const float* __restrict__ b2, float* __restrict__ out) {
  int g = blockIdx.x * blockDim.x + threadIdx.x;
  if (g >= N_GRAPHS) return;
  float logits[N_CLASSES];
#pragma unroll
  for (int c = 0; c < N_CLASSES; ++c) logits[c] = b2[c];
  for (int k = 0; k < DIM; ++k) {
    float hv = ph[(size_t)g * DIM + k];
#pragma unroll
    for (int c = 0; c < N_CLASSES; ++c) logits[c] += hv * W2[k * N_CLASSES + c];
  }
  float mx = logits[0];
#pragma unroll
  for (int c = 1; c < N_CLASSES; ++c) mx = logits[c] > mx ? logits[c] : mx;
  float sum = 0.0f;
#pragma unroll
  for (int c = 0; c < N_CLASSES; ++c) sum += __expf(logits[c] - mx);
  float lse = mx + __logf(sum);
#pragma unroll
  for (int c = 0; c < N_CLASSES; ++c) out[(size_t)g * N_CLASSES + c] = logits[c] - lse;
}

// ---------------------------------------------------------------- driver
extern "C" void kernel_launch(void* const* d_in, const int* in_sizes, int n_in,
                              void* d_out, int out_size, void* d_ws, size_t ws_size,
                              hipStream_t stream) {
  const float* x      = (const float*)d_in[0];
  const int*   ei     = (const int*)d_in[1];
  const int*   batch  = (const int*)d_in[2];
  const float* W1s    = (const float*)d_in[3];
  const float* b1s    = (const float*)d_in[4];
  const float* W2s    = (const float*)d_in[5];
  const float* b2s    = (const float*)d_in[6];
  const float* gammas = (const float*)d_in[7];
  const float* betas  = (const float*)d_in[8];
  const float* eps    = (const float*)d_in[9];
  const float* fc1_W  = (const float*)d_in[10];
  const float* fc1_b  = (const float*)d_in[11];
  const float* fc2_W  = (const float*)d_in[12];
  const float* fc2_b  = (const float*)d_in[13];

  const int* src = ei;             // edge_index[0]
  const int* dst = ei + N_EDGES;   // edge_index[1]

  float* bufA    = (float*)d_ws;           // [N_NODES*DIM]
  float* bufB    = bufA + ND;              // [N_NODES*DIM]
  float* stats   = bufB + ND;              // [256]  (sum | sumsq)
  float* psums   = stats + 256;            // [128*128]
  float* pcnts   = psums + N_GRAPHS * DIM; // [128]
  float* pooled  = pcnts + N_GRAPHS;       // [128*128]
  float* pooledH = pooled + N_GRAPHS * DIM;// [128*128]
  float* Wt      = pooledH + N_GRAPHS*DIM; // [128*128] transposed weight scratch

  const int rowTiles = N_NODES / 16;       // 6250, exact
  const int elem4    = (int)(ND / 4);      // 3.2M float4
  const int edgeBlks = (N_EDGES + 7) / 8;  // 8 waves per 256-thread block
  const int tBlks    = (DIM * DIM) / 256;  // 64

  const float* xin = x;
  for (int l = 0; l < N_LAYERS; ++l) {
    // agg = (1+eps)*x + scatter_add(x[src] -> dst)
    gin_init<<<elem4 / 256, 256, 0, stream>>>(xin, bufA, eps, l);
    gin_scatter<<<edgeBlks, 256, 0, stream>>>(xin, src, dst, bufA, N_EDGES);
    // MLP: Linear-ReLU-Linear-ReLU   (WMMA f32 16x16x4, weights pre-transposed)
    transpose128<<<tBlks, 256, 0, stream>>>(W1s + (size_t)l * DIM * DIM, Wt);
    gemm_bias_relu<<<rowTiles, 256, 0, stream>>>(bufA, Wt, b1s + l * DIM, bufB);
    transpose128<<<tBlks, 256, 0, stream>>>(W2s + (size_t)l * DIM * DIM, Wt);
    gemm_bias_relu<<<rowTiles, 256, 0, stream>>>(bufB, Wt, b2s + l * DIM, bufA);
    // BatchNorm (training-mode batch stats)
    zero_f32<<<1, 256, 0, stream>>>(stats, 256);
    bn_reduce<<<N_NODES / BN_ROWS, DIM, 0, stream>>>(bufA, stats, N_NODES);
    bn_norm<<<(int)(ND / 256), 256, 0, stream>>>(bufA, stats, gammas + l * DIM,
                                                 betas + l * DIM, bufB, N_NODES);
    xin = bufB;
  }

  // global mean pool  (psums and pcnts are contiguous -> one zero pass)
  zero_f32<<<(N_GRAPHS * DIM + N_GRAPHS + 255) / 256, 256, 0, stream>>>(psums, N_GRAPHS * DIM + N_GRAPHS);
  pool_scatter<<<(N_NODES + 7) / 8, 256, 0, stream>>>(bufB, batch, psums, pcnts, N_NODES);
  pool_div<<<(N_GRAPHS * DIM + 255) / 256, 256, 0, stream>>>(psums, pcnts, pooled);

  // fc1 (+ReLU) via WMMA, then fc2 + log_softmax
  transpose128<<<tBlks, 256, 0, stream>>>(fc1_W, Wt);
  gemm_bias_relu<<<N_GRAPHS / 16, 256, 0, stream>>>(pooled, Wt, fc1_b, pooledH);
  head_kernel<<<1, N_GRAPHS, 0, stream>>>(pooledH, fc2_W, fc2_b, (float*)d_out);
}